// WarpingLayer_51221779972862
// MI455X (gfx1250) — compile-verified
//
#include <hip/hip_runtime.h>
#include <hip/hip_bf16.h>
#include <stdint.h>

// Problem shape (fixed by reference setup_inputs): x[8,128,128,256] f32, flow[8,2,128,256] f32.
constexpr int B   = 8;
constexpr int C   = 128;
constexpr int H   = 128;
constexpr int W   = 256;
constexpr int HW  = H * W;       // 32768 = 2^15
constexpr int CHW = C * HW;
constexpr int CPT = 32;          // channels per thread (blockIdx.y splits C into C/CPT chunks)

#if __has_builtin(__builtin_amdgcn_global_load_async_to_lds_b32)
#define HAS_ASYNC_LDS 1
#else
#define HAS_ASYNC_LDS 0
#endif

typedef __attribute__((address_space(1))) int* gptr_i32;
typedef __attribute__((address_space(3))) int* lptr_i32;

__global__ __launch_bounds__(256) void warp_bilinear_kernel(
    const float* __restrict__ x,
    const float* __restrict__ flow,
    float* __restrict__ out) {

  const int t   = threadIdx.x;
  const int pix = blockIdx.x * 256 + t;     // [0, B*H*W)
  const int b   = pix >> 15;                // / HW
  const int hw  = pix & (HW - 1);
  const int h   = hw >> 8;                  // / W
  const int w   = hw & (W - 1);

  // ---- Stage this pixel's flow (fx, fy) via async global->LDS (CDNA5 path) ----
  __shared__ float sflow[2 * 256];
  const float* gf = flow + (size_t)b * (2 * HW) + hw;   // channel 0; channel 1 at +HW
  float fx, fy;
#if HAS_ASYNC_LDS
  {
    // generic->AS1 / AS3 via integer round-trip (low 32 bits of a generic LDS
    // pointer are the LDS byte offset on AMDGPU). Builtin takes AS1/AS3 int*.
    gptr_i32 g0 = (gptr_i32)(uintptr_t)gf;
    lptr_i32 l0 = (lptr_i32)(uint32_t)(uintptr_t)&sflow[t];
    lptr_i32 l1 = (lptr_i32)(uint32_t)(uintptr_t)&sflow[256 + t];
    __builtin_amdgcn_global_load_async_to_lds_b32(g0, l0, 0, 0);
    __builtin_amdgcn_global_load_async_to_lds_b32(g0 + HW, l1, 0, 0);
#if __has_builtin(__builtin_amdgcn_s_wait_asynccnt)
    __builtin_amdgcn_s_wait_asynccnt(0);
#else
    asm volatile("s_wait_asynccnt 0" ::: "memory");
#endif
    // Each lane reads only the slots it wrote, so a per-wave async wait suffices.
    fx = sflow[t];
    fy = sflow[256 + t];
  }
#else
  fx = __builtin_nontemporal_load(gf);
  fy = __builtin_nontemporal_load(gf + HW);
#endif

  // ---- Per-pixel coordinates & bilinear weights (reference arithmetic replicated) ----
  constexpr float Wm1 = (float)(W - 1);
  constexpr float Hm1 = (float)(H - 1);
  float ix = (float)w + fx;
  float iy = (float)h + fy;
  float ixn = 2.0f * ix / Wm1 - 1.0f;       // normalize (align_corners=True)
  float iyn = 2.0f * iy / Hm1 - 1.0f;
  float ixa = (ixn + 1.0f) * 0.5f * Wm1;    // unnormalize (not an fp32 identity!)
  float iya = (iyn + 1.0f) * 0.5f * Hm1;

  float x0f = floorf(ixa), y0f = floorf(iya);
  float x1f = x0f + 1.0f,  y1f = y0f + 1.0f;
  float wx1 = ixa - x0f,   wx0 = 1.0f - wx1;
  float wy1 = iya - y0f,   wy0 = 1.0f - wy1;

  // zeros padding: validity mask folded into the tap weight
  float vx0 = (x0f >= 0.0f && x0f <= Wm1) ? 1.0f : 0.0f;
  float vx1 = (x1f >= 0.0f && x1f <= Wm1) ? 1.0f : 0.0f;
  float vy0 = (y0f >= 0.0f && y0f <= Hm1) ? 1.0f : 0.0f;
  float vy1 = (y1f >= 0.0f && y1f <= Hm1) ? 1.0f : 0.0f;

  float w00 = wx0 * wy0 * vx0 * vy0;
  float w10 = wx1 * wy0 * vx1 * vy0;
  float w01 = wx0 * wy1 * vx0 * vy1;
  float w11 = wx1 * wy1 * vx1 * vy1;

  int X0 = (int)fminf(fmaxf(x0f, 0.0f), Wm1);
  int X1 = (int)fminf(fmaxf(x1f, 0.0f), Wm1);
  int Y0 = (int)fminf(fmaxf(y0f, 0.0f), Hm1);
  int Y1 = (int)fminf(fmaxf(y1f, 0.0f), Hm1);

  const int o00 = Y0 * W + X0;
  const int o10 = Y0 * W + X1;
  const int o01 = Y1 * W + X0;
  const int o11 = Y1 * W + X1;

  // ---- Channel loop: same 4 offsets & weights for every channel plane ----
  const int    c0 = blockIdx.y * CPT;
  const float* px = x   + (size_t)b * CHW + (size_t)c0 * HW;
  float*       po = out + (size_t)b * CHW + (size_t)c0 * HW + hw;

#pragma unroll 4
  for (int c = 0; c < CPT; ++c) {
    if (c + 4 < CPT) {
      // global_prefetch_b8 on the two tap-row streams, 4 channel planes ahead
      __builtin_prefetch(px + 4 * HW + o00, 0, 0);
      __builtin_prefetch(px + 4 * HW + o01, 0, 0);
    }
    float v00 = px[o00];
    float v10 = px[o10];
    float v01 = px[o01];
    float v11 = px[o11];
    float r = ((w00 * v00 + w10 * v10) + w01 * v01) + w11 * v11;
    // NT store: out is write-once (128 MB) — keep it from evicting x (128 MB)
    // out of the 192 MB L2, preserving the 4-tap gather reuse.
    __builtin_nontemporal_store(r, po);
    px += HW;
    po += HW;
  }
}

extern "C" void kernel_launch(void* const* d_in, const int* in_sizes, int n_in,
                              void* d_out, int out_size, void* d_ws, size_t ws_size,
                              hipStream_t stream) {
  const float* x    = (const float*)d_in[0];
  const float* flow = (const float*)d_in[1];
  float*       out  = (float*)d_out;
  (void)in_sizes; (void)n_in; (void)out_size; (void)d_ws; (void)ws_size;

  dim3 grid((B * H * W) / 256, C / CPT);   // 1024 x 4 blocks
  warp_bilinear_kernel<<<grid, dim3(256), 0, stream>>>(x, flow, out);
}